// LocalConvolutionMix_75325136437781
// MI455X (gfx1250) — compile-verified
//
#include <hip/hip_runtime.h>
#include <cstdint>

// ---------------------------------------------------------------------------
// Involution-style dual local conv (3x3 pad1 + 5x5 pad2), fp32, memory-bound.
// N=4, C=256, H=W=56, heads=1, wc=32  ->  G = 8 channel groups share weights.
// Roofline: ~218 MFLOP vs ~93 MB unique traffic -> 2.3 FLOP/B, bound by HBM
// (23.3 TB/s -> ~4us floor). WMMA cannot map (per-pixel weights, contraction
// batched per pixel). CDNA5 paths used: async global->LDS (ASYNCcnt) double-
// buffered pipeline + non-temporal hints on the single-use weight/out streams.
// ---------------------------------------------------------------------------

#define N_  4
#define C_  256
#define H_  56
#define W_  56
#define WC_ 32
#define G_  8
#define HW_ (H_ * W_)

struct alignas(16) F4 { float x, y, z, w; };
typedef float f2v __attribute__((ext_vector_type(2)));

typedef int i4v __attribute__((vector_size(16)));
typedef __attribute__((address_space(1))) i4v* gas_i4p;
typedef __attribute__((address_space(3))) i4v* las_i4p;

#if defined(__AMDGCN__) && __has_builtin(__builtin_amdgcn_global_load_async_to_lds_b128)
#define HAVE_ASYNC_LDS 1
#else
#define HAVE_ASYNC_LDS 0
#endif

// 16-byte global -> LDS copy. Async (ASYNCcnt-tracked) when available.
__device__ __forceinline__ void stage16(const float* __restrict__ g, float* l) {
#if HAVE_ASYNC_LDS
  __builtin_amdgcn_global_load_async_to_lds_b128(
      (gas_i4p)(unsigned long long)(uintptr_t)g,
      (las_i4p)(unsigned int)(uintptr_t)l,
      0, 0);
#else
  *(F4*)l = *(const F4*)g;
#endif
}

__device__ __forceinline__ void wait_async_all() {
#if HAVE_ASYNC_LDS
#if __has_builtin(__builtin_amdgcn_s_wait_asynccnt)
  __builtin_amdgcn_s_wait_asynccnt(0);
#else
  asm volatile("s_wait_asynccnt 0" ::: "memory");
#endif
#endif
}

__global__ __launch_bounds__(224)
void involution_mix_kernel(const float* __restrict__ x,
                           const float* __restrict__ w1,
                           const float* __restrict__ w2,
                           float* __restrict__ out) {
  // LDS x tile: [buf][ch 0..7][row 0..4][col 0..63]; interior cols at idx 4..59
  // (global col c -> lds idx c+4), halo idx {0..3, 60..63} stay zero.
  __shared__ __align__(16) float tile[2][8][5][64];

  const int tx    = threadIdx.x;            // 0..27  : pair of output columns
  const int ty    = threadIdx.y;            // 0..7   : local weight lane / ch slot
  const int tid   = ty * 28 + tx;           // 0..223
  const int nh    = blockIdx.x;             // n*56 + h
  const int n     = nh / H_;
  const int h     = nh % H_;
  const int vbase = blockIdx.y * 8;         // weight-lane group base
  const int v     = vbase + ty;             // 0..31
  const int w0    = tx * 2;                 // even output column

  float* const lds = (float*)tile;

  // ---- one-time zero of halo columns (shift-only addressing) --------------
  // 2*8*5 = 80 LDS rows, 8 halo elements each (idx 0..3 and 60..63).
  for (int q = tid; q < 640; q += 224) {
    const int rowid = q >> 3;
    const int j     = q & 7;
    lds[(rowid << 6) + ((j < 4) ? j : 56 + j)] = 0.0f;
  }
  // ---- one-time zero of out-of-range image rows (uniform branch) ----------
  if (h < 2 || h > H_ - 3) {
#pragma unroll
    for (int r = 0; r < 5; ++r) {
      const int hh = h - 2 + r;
      if (hh < 0 || hh >= H_) {
        for (int q = tid; q < 1024; q += 224) {       // 16 (b,ch) * 64 idx
          const int bch = q >> 6;
          const int idx = q & 63;
          if (idx >= 4 && idx < 60) lds[bch * 320 + (r << 6) + idx] = 0.0f;
        }
      }
    }
  }

  // ---- preload per-pixel weights: read each weight exactly once (NT) ------
  f2v w2r[25], w1r[9];
  {
    const size_t b2 = ((size_t)(n * WC_ + v) * 25) * HW_ + (size_t)h * W_ + w0;
#pragma unroll
    for (int t = 0; t < 25; ++t)
      w2r[t] = __builtin_nontemporal_load((const f2v*)(w2 + b2 + (size_t)t * HW_));
    const size_t b1 = ((size_t)(n * WC_ + v) * 9) * HW_ + (size_t)h * W_ + w0;
#pragma unroll
    for (int t = 0; t < 9; ++t)
      w1r[t] = __builtin_nontemporal_load((const f2v*)(w1 + b1 + (size_t)t * HW_));
  }

  // ---- per-thread staging slots (g-invariant, hoisted) --------------------
  // 8 ch * 5 rows * 14 float4 = 560 b128 transfers per block per group.
  const float* sg[3];     // global source for g=0 (advance by WC_*HW_ per g)
  float*       sl0[3];    // LDS destination, buffer 0
  float*       sl1[3];    // LDS destination, buffer 1
  bool         sa[3];     // slot active
#pragma unroll
  for (int it = 0; it < 3; ++it) {
    const int q  = tid + it * 224;
    int ch  = q / 70;  if (ch > 7) ch = 7;
    const int r   = q % 70;
    const int row = r / 14;
    const int c4  = r % 14;
    const int hh  = h - 2 + row;
    const bool act = (q < 560) && (hh >= 0) && (hh < H_);
    sa[it]  = act;
    sg[it]  = x + ((size_t)(n * C_ + vbase + ch) * H_ + (act ? hh : 0)) * W_ + c4 * 4;
    sl0[it] = &tile[0][ch][row][4 + c4 * 4];
    sl1[it] = &tile[1][ch][row][4 + c4 * 4];
  }

  // prefetch group 0 into buffer 0
#pragma unroll
  for (int it = 0; it < 3; ++it)
    if (sa[it]) stage16(sg[it], sl0[it]);

#pragma unroll
  for (int g = 0; g < G_; ++g) {
    wait_async_all();      // my wave's async copies for buf[g&1] are in LDS
    __syncthreads();       // whole block: tile ready, previous compute done

    if (g + 1 < G_) {      // prefetch next group into the other buffer
      const size_t goff = (size_t)(g + 1) * (WC_ * HW_);
#pragma unroll
      for (int it = 0; it < 3; ++it)
        if (sa[it]) stage16(sg[it] + goff, ((g + 1) & 1) ? sl1[it] : sl0[it]);
    }

    const int buf = g & 1;
    float a1x = 0.f, a1y = 0.f, a2x = 0.f, a2y = 0.f;

#pragma unroll
    for (int row = 0; row < 5; ++row) {
      const float* lr = &tile[buf][ty][row][0];
      float xr[6];
      { f2v t0 = *(const f2v*)&lr[w0 + 2]; xr[0] = t0.x; xr[1] = t0.y; }
      { f2v t1 = *(const f2v*)&lr[w0 + 4]; xr[2] = t1.x; xr[3] = t1.y; }
      { f2v t2 = *(const f2v*)&lr[w0 + 6]; xr[4] = t2.x; xr[5] = t2.y; }
#pragma unroll
      for (int d = 0; d < 5; ++d) {
        const f2v wt = w2r[row * 5 + d];
        a2x = fmaf(wt.x, xr[d],     a2x);
        a2y = fmaf(wt.y, xr[d + 1], a2y);
        if (row >= 1 && row <= 3 && d >= 1 && d <= 3) {
          const f2v ws = w1r[(row - 1) * 3 + (d - 1)];
          a1x = fmaf(ws.x, xr[d],     a1x);
          a1y = fmaf(ws.y, xr[d + 1], a1y);
        }
      }
    }

    const int c = g * WC_ + v;
    const size_t o1 = ((size_t)((n * 2 + 0) * C_ + c)) * HW_ + (size_t)h * W_ + w0;
    const size_t o2 = ((size_t)((n * 2 + 1) * C_ + c)) * HW_ + (size_t)h * W_ + w0;
    f2v r1 = {a1x, a1y};
    f2v r2 = {a2x, a2y};
    __builtin_nontemporal_store(r1, (f2v*)(out + o1));   // write-once stream
    __builtin_nontemporal_store(r2, (f2v*)(out + o2));
  }
}

extern "C" void kernel_launch(void* const* d_in, const int* in_sizes, int n_in,
                              void* d_out, int out_size, void* d_ws, size_t ws_size,
                              hipStream_t stream) {
  (void)in_sizes; (void)n_in; (void)out_size; (void)d_ws; (void)ws_size;
  const float* x  = (const float*)d_in[0];
  const float* w1 = (const float*)d_in[1];
  const float* w2 = (const float*)d_in[2];
  float* out      = (float*)d_out;

  dim3 grid(N_ * H_, 4, 1);   // 224 (n,h) blocks * 4 v-groups = 896 blocks
  dim3 block(28, 8, 1);       // 224 threads = 7 wave32
  hipLaunchKernelGGL(involution_mix_kernel, grid, block, 0, stream, x, w1, w2, out);
}